// EnergySRB_14680198218405
// MI455X (gfx1250) — compile-verified
//
#include <hip/hip_runtime.h>
#include <hip/hip_bf16.h>
#include <stdint.h>

// ---------------------------------------------------------------------------
// EnergySRB on MI455X (gfx1250, wave32)
//
//   srb(p) = pre[s0,s1] * exp(dfac[s0,s1]*d + 1 - 1/(1-(d/rc)^2))   (d < rc)
//   out[mol] = energies[mol] + sum_p srb(p),  mol = idx0[p]/n_atoms
//
// Strategy:
//   * species packed to 2 bits/atom in d_ws (64KB -> WGP$-resident gathers)
//   * 4x4 {pre,dfac} tables staged as float2 in LDS
//   * streamed idx0/idx1/dist tiles double-buffered into LDS with
//     global_load_async_to_lds_b128 (+ s_wait_asynccnt), 16B per thread per array
//   * per-block LDS accumulator acc[4096] with ds_add_f32, single global
//     atomic flush per (block, mol)
// ---------------------------------------------------------------------------

#define A2B 1.8897261258369282f
#define RC  (5.2f * A2B)

constexpr int THREADS = 256;
constexpr int VEC     = 4;
constexpr int TILE    = THREADS * VEC;     // 1024 pairs / tile
constexpr int LDS_MOLS = 4096;

// ---------------------------------------------------------------------------
// Kernel 1: seed d_out (tuple layout: [species int32 ...][energies float ...])
// ---------------------------------------------------------------------------
__global__ void init_out_kernel(const int* __restrict__ species,
                                const float* __restrict__ energies,
                                int* __restrict__ outSpecies,
                                float* __restrict__ outE,
                                int nSpecies, int nMol, int copySpecies) {
    int i = blockIdx.x * blockDim.x + threadIdx.x;
    if (copySpecies && i < nSpecies) outSpecies[i] = species[i];
    if (i < nMol) outE[i] = energies[i];
}

// ---------------------------------------------------------------------------
// Kernel 2: pack species (values 0..3) into 2 bits/atom -> 64KB table
// ---------------------------------------------------------------------------
__global__ void pack_species_kernel(const int* __restrict__ species,
                                    unsigned char* __restrict__ packed,
                                    int nBytes, int nTotal) {
    int j = blockIdx.x * blockDim.x + threadIdx.x;
    if (j >= nBytes) return;
    unsigned v = 0u;
#pragma unroll
    for (int k = 0; k < 4; ++k) {
        int idx = 4 * j + k;
        int s = (idx < nTotal) ? species[idx] : 0;
        v |= ((unsigned)s & 3u) << (2 * k);
    }
    packed[j] = (unsigned char)v;
}

// ---------------------------------------------------------------------------
// Kernel 3: main pair loop
// ---------------------------------------------------------------------------
__global__ __launch_bounds__(THREADS) void srb_kernel(
    const int* __restrict__ idx0, const int* __restrict__ idx1,
    const float* __restrict__ dist,
    const unsigned char* __restrict__ packed,   // may be nullptr
    const int* __restrict__ speciesFull,
    const float* __restrict__ expPre, const float* __restrict__ dFac,
    float* __restrict__ outE,
    int nPairs, int nElem, int nMol, int atomsShift, int nAtoms, int numTiles) {

    __shared__ float  acc[LDS_MOLS];
    __shared__ float2 tab[64];
    __shared__ int4   sI0[2][THREADS];
    __shared__ int4   sI1[2][THREADS];
    __shared__ float4 sD[2][THREADS];

    const int tid = threadIdx.x;

    for (int m = tid; m < LDS_MOLS; m += THREADS) acc[m] = 0.f;
    if (tid < nElem * nElem && tid < 64)
        tab[tid] = make_float2(expPre[tid], dFac[tid]);
    __syncthreads();

    const float invRC    = 1.0f / RC;
    const bool  useLds   = (nMol <= LDS_MOLS);
    const bool  usePack  = (packed != nullptr);

    // stage one tile's worth of idx0/idx1/dist (16B each per thread) into LDS
    // asynchronously.  Addresses are clamped+aligned so the issue count per
    // wave is always exactly 3 (keeps scalar s_wait_asynccnt bookkeeping valid)
    auto stage = [&](int tile, int buf) {
        long b = (long)tile * TILE + (long)tid * VEC;
        long c = b;
        long maxb = (long)nPairs - VEC;
        if (c > maxb) c = maxb;
        if (c < 0)    c = 0;
        c &= ~(long)(VEC - 1);   // keep 16B alignment for b128
        unsigned l0 = (unsigned)(size_t)(&sI0[buf][tid]);
        unsigned l1 = (unsigned)(size_t)(&sI1[buf][tid]);
        unsigned l2 = (unsigned)(size_t)(&sD[buf][tid]);
        unsigned long long g0 = (unsigned long long)(size_t)(idx0 + c);
        unsigned long long g1 = (unsigned long long)(size_t)(idx1 + c);
        unsigned long long g2 = (unsigned long long)(size_t)(dist + c);
        asm volatile("global_load_async_to_lds_b128 %0, %1, off"
                     :: "v"(l0), "v"(g0) : "memory");
        asm volatile("global_load_async_to_lds_b128 %0, %1, off"
                     :: "v"(l1), "v"(g1) : "memory");
        asm volatile("global_load_async_to_lds_b128 %0, %1, off"
                     :: "v"(l2), "v"(g2) : "memory");
    };

    auto process = [&](int tile, int buf) {
        long base = (long)tile * TILE + (long)tid * VEC;
        bool full = (base + VEC <= (long)nPairs);
        int   i0v[VEC], i1v[VEC];
        float dv[VEC];
        if (full) {
            int4   a0 = sI0[buf][tid];          // ds_load_b128
            int4   a1 = sI1[buf][tid];
            float4 dd = sD[buf][tid];
            i0v[0] = a0.x; i0v[1] = a0.y; i0v[2] = a0.z; i0v[3] = a0.w;
            i1v[0] = a1.x; i1v[1] = a1.y; i1v[2] = a1.z; i1v[3] = a1.w;
            dv[0] = dd.x; dv[1] = dd.y; dv[2] = dd.z; dv[3] = dd.w;
        }
#pragma unroll
        for (int j = 0; j < VEC; ++j) {
            long p = base + j;
            if (!full) {                         // rare tail: direct loads
                if (p >= (long)nPairs) continue;
                i0v[j] = idx0[p]; i1v[j] = idx1[p]; dv[j] = dist[p];
            }
            const int   i0 = i0v[j];
            const int   i1 = i1v[j];
            const float d  = dv[j] * A2B;
            if (d < RC) {
                int s0, s1;
                if (usePack) {
                    unsigned b0 = packed[(unsigned)i0 >> 2];
                    unsigned b1 = packed[(unsigned)i1 >> 2];
                    s0 = (int)((b0 >> (((unsigned)i0 & 3u) << 1)) & 3u);
                    s1 = (int)((b1 >> (((unsigned)i1 & 3u) << 1)) & 3u);
                } else {
                    s0 = speciesFull[i0];
                    s1 = speciesFull[i1];
                }
                float2 pf = tab[s0 * nElem + s1];
                float x  = d * invRC;
                float om = 1.0f - x * x;
                float e  = pf.y * d + 1.0f - 1.0f / om;
                float srb = pf.x * __expf(e);
                int mol = (atomsShift >= 0) ? (int)((unsigned)i0 >> atomsShift)
                                            : (int)((unsigned)i0 / (unsigned)nAtoms);
                if (useLds) atomicAdd(&acc[mol], srb);   // ds_add_f32
                else        atomicAdd(&outE[mol], srb);
            }
        }
    };

    // grid-stride over tiles with double-buffered async staging
    int t = blockIdx.x;
    int cur = 0;
    if (t < numTiles) stage(t, 0);
    for (; t < numTiles; t += gridDim.x) {
        int nt = t + gridDim.x;
        if (nt < numTiles) {
            stage(nt, cur ^ 1);
            asm volatile("s_wait_asynccnt 0x3" ::: "memory");  // current buf done
        } else {
            asm volatile("s_wait_asynccnt 0x0" ::: "memory");
        }
        process(t, cur);
        cur ^= 1;
    }

    __syncthreads();
    if (useLds) {
        for (int m = tid; m < nMol; m += THREADS) {
            float v = acc[m];
            if (v != 0.0f) atomicAdd(&outE[m], v);
        }
    }
}

// ---------------------------------------------------------------------------
extern "C" void kernel_launch(void* const* d_in, const int* in_sizes, int n_in,
                              void* d_out, int out_size, void* d_ws, size_t ws_size,
                              hipStream_t stream) {
    const int*   species  = (const int*)d_in[0];   int nSpecies = in_sizes[0];
    const float* energies = (const float*)d_in[1]; int nMol     = in_sizes[1];
    const int*   atomIdx  = (const int*)d_in[2];
    const float* dist     = (const float*)d_in[3]; int nPairs   = in_sizes[3];
    const float* expPre   = (const float*)d_in[4]; int nElem2   = in_sizes[4];
    const float* dFac     = (const float*)d_in[5];

    int nElem = 1;
    while (nElem * nElem < nElem2) ++nElem;
    int nAtoms = (nMol > 0) ? nSpecies / nMol : 1;
    if (nAtoms <= 0) nAtoms = 1;

    const int* idx0 = atomIdx;
    const int* idx1 = atomIdx + nPairs;

    // tuple output: [species (nSpecies int32)] [out (nMol float)]
    bool tupleOut = (out_size >= nSpecies + nMol);
    int*   outSpecies = (int*)d_out;
    float* outE = tupleOut ? ((float*)d_out + nSpecies) : (float*)d_out;

    int initN = tupleOut ? (nSpecies > nMol ? nSpecies : nMol) : nMol;
    init_out_kernel<<<(initN + THREADS - 1) / THREADS, THREADS, 0, stream>>>(
        species, energies, outSpecies, outE, nSpecies, nMol, tupleOut ? 1 : 0);

    // 2-bit species pack into workspace (fits WGP$)
    const unsigned char* packedPtr = nullptr;
    int nBytes = (nSpecies + 3) / 4;
    if (nElem <= 4 && ws_size >= (size_t)nBytes && nBytes > 0) {
        unsigned char* pk = (unsigned char*)d_ws;
        pack_species_kernel<<<(nBytes + THREADS - 1) / THREADS, THREADS, 0, stream>>>(
            species, pk, nBytes, nSpecies);
        packedPtr = pk;
    }

    int atomsShift = ((nAtoms & (nAtoms - 1)) == 0) ? __builtin_ctz((unsigned)nAtoms) : -1;
    int numTiles = (int)(((long)nPairs + TILE - 1) / TILE);
    if (numTiles <= 0) return;
    int numBlocks = numTiles < 512 ? numTiles : 512;

    srb_kernel<<<numBlocks, THREADS, 0, stream>>>(
        idx0, idx1, dist, packedPtr, species, expPre, dFac, outE,
        nPairs, nElem, nMol, atomsShift, nAtoms, numTiles);
}